// BlockGNN_28123445854283
// MI455X (gfx1250) — compile-verified
//
#include <hip/hip_runtime.h>
#include <hip/hip_bf16.h>

// BlockGNN for MI455X (gfx1250). All matmuls via V_WMMA_F32_16X16X4_F32
// (exact fp32). Weights are pre-packed into B-fragment order so every lane
// fetches its v2f fragment with one coalesced global_load_b64. Each wave
// computes a 32-row x 128-col tile (two 16x128 C halves) so each B fragment
// feeds two WMMAs. Edge streams (edge_attr in / messages out, ~307MB/layer)
// use non-temporal policy; node table + agg (51MB each) stay L2-resident.

#define D 128

typedef __attribute__((ext_vector_type(2))) float v2f;
typedef __attribute__((ext_vector_type(8))) float v8f;

__global__ void zero4_kernel(float4* p, size_t n4) {
    size_t i = (size_t)blockIdx.x * blockDim.x + threadIdx.x;
    size_t stride = (size_t)gridDim.x * blockDim.x;
    float4 z = make_float4(0.f, 0.f, 0.f, 0.f);
    for (; i < n4; i += stride) p[i] = z;
}

// Repack W[K][128] (row-major) into WMMA B-fragment order:
// dst[g*512 + n*64 + lane*2 + half] = W[g*4 + 2*(lane>>4) + half][n*16 + (lane&15)]
// so a lane's v2f fragment for (k-group g, n-tile n) is one contiguous b64.
__global__ void pack_wfrag_kernel(const float* __restrict__ src,
                                  float* __restrict__ dst, int total) {
    for (int t = blockIdx.x * blockDim.x + threadIdx.x; t < total;
         t += gridDim.x * blockDim.x) {
        int g    = t >> 9;
        int rem  = t & 511;
        int n    = rem >> 6;
        int l2   = rem & 63;
        int lane = l2 >> 1;
        int half = l2 & 1;
        int k    = (g << 2) + ((lane >> 4) << 1) + half;
        int col  = n * 16 + (lane & 15);
        dst[t] = src[k * D + col];
    }
}

// out_row = relu(concat(seg_0..seg_{NSEG-1})[row] @ W1 + b1) @ W2 + b2
// One wave per 32-row tile. W1f/W2f are fragment-packed.
// NTIN: non-temporal loads for direct (non-gathered) segments.
// NTOUT: non-temporal stores for out.
template <int NSEG, bool SCATTER, bool NTIN, bool NTOUT>
__global__ __launch_bounds__(64, 1) void mlp2_wmma_kernel(
    const float* s0, const int* i0,
    const float* s1, const int* i1,
    const float* s2, const int* i2,
    const float* __restrict__ W1f, const float* __restrict__ b1,
    const float* __restrict__ W2f, const float* __restrict__ b2,
    float* out, const int* sidx, float* agg, int rows)
{
    __shared__ float hid[2][32 * 132];     // per-wave 32x128 tile, stride 132

    const int lane   = threadIdx.x & 31;
    const int wave   = threadIdx.x >> 5;
    const int r      = lane & 15;          // A row / B,C col within 16-tile
    const int kh     = (lane >> 4) << 1;   // K offset 0/2 for A fragments
    const int rowoff = (lane >> 4) << 3;   // M offset 0/8 for C fragments

    const int tile     = blockIdx.x * 2 + wave;
    const int row_base = tile * 32;
    if (row_base >= rows) return;          // safe: no block-wide barrier used
    const int row0 = row_base + r;
    const int row1 = row_base + 16 + r;

    const float* segp0[NSEG];
    const float* segp1[NSEG];
    bool segnt[NSEG];
    {
        const float* sp[3] = { s0, s1, s2 };
        const int*   ip[3] = { i0, i1, i2 };
        #pragma unroll
        for (int s = 0; s < NSEG; ++s) {
            int ra = ip[s] ? ip[s][row0] : row0;
            int rb = ip[s] ? ip[s][row1] : row1;
            segp0[s] = sp[s] + (size_t)ra * D;
            segp1[s] = sp[s] + (size_t)rb * D;
            segnt[s] = NTIN && (ip[s] == nullptr);
        }
    }

    // ---- GEMM 1: [32 x NSEG*128] @ [NSEG*128 x 128] ----
    v8f accL[8] = {};                      // rows row_base .. +15
    v8f accH[8] = {};                      // rows row_base+16 .. +31
    #pragma unroll
    for (int s = 0; s < NSEG; ++s) {
        const float* a0p = segp0[s];
        const float* a1p = segp1[s];
        const bool nt = segnt[s];
        const float* wf = W1f + (size_t)(s * 32) * 512 + lane * 2;
        for (int ks = 0; ks < 32; ++ks) {
            const int k = ks * 4 + kh;
            v2f a0 = nt ? __builtin_nontemporal_load((const v2f*)(a0p + k))
                        : *(const v2f*)(a0p + k);
            v2f a1 = nt ? __builtin_nontemporal_load((const v2f*)(a1p + k))
                        : *(const v2f*)(a1p + k);
            const float* wk = wf + (size_t)ks * 512;
            #pragma unroll
            for (int n = 0; n < 8; ++n) {
                v2f b = *(const v2f*)(wk + n * 64);   // coalesced 256B/instr
                accL[n] = __builtin_amdgcn_wmma_f32_16x16x4_f32(
                    false, a0, false, b, (short)0, accL[n], false, false);
                accH[n] = __builtin_amdgcn_wmma_f32_16x16x4_f32(
                    false, a1, false, b, (short)0, accH[n], false, false);
            }
        }
    }

    // bias + ReLU, park hidden tile in LDS (C layout -> row-major)
    float* h = &hid[wave][0];
    #pragma unroll
    for (int n = 0; n < 8; ++n) {
        const float bb = b1[n * 16 + r];
        #pragma unroll
        for (int j = 0; j < 8; ++j) {
            float vL = accL[n][j] + bb;
            float vH = accH[n][j] + bb;
            vL = vL > 0.f ? vL : 0.f;
            vH = vH > 0.f ? vH : 0.f;
            h[(j + rowoff) * 132 + n * 16 + r]        = vL;
            h[(j + rowoff + 16) * 132 + n * 16 + r]   = vH;
        }
    }
    // wave-local LDS RAW: DS ops in-order per wave; compiler inserts s_wait_dscnt.

    // ---- GEMM 2: [32 x 128] @ [128 x 128] ----
    v8f cL[8] = {};
    v8f cH[8] = {};
    for (int ks = 0; ks < 32; ++ks) {
        const int k = ks * 4 + kh;
        v2f a0 = *(const v2f*)&h[r * 132 + k];
        v2f a1 = *(const v2f*)&h[(r + 16) * 132 + k];
        const float* wk = W2f + (size_t)ks * 512 + lane * 2;
        #pragma unroll
        for (int n = 0; n < 8; ++n) {
            v2f b = *(const v2f*)(wk + n * 64);
            cL[n] = __builtin_amdgcn_wmma_f32_16x16x4_f32(
                false, a0, false, b, (short)0, cL[n], false, false);
            cH[n] = __builtin_amdgcn_wmma_f32_16x16x4_f32(
                false, a1, false, b, (short)0, cH[n], false, false);
        }
    }

    // bias + store (+ optional scatter-add into agg)
    float b2v[8];
    #pragma unroll
    for (int n = 0; n < 8; ++n) b2v[n] = b2[n * 16 + r];

    #pragma unroll
    for (int j = 0; j < 8; ++j) {
        const int rowA = row_base + j + rowoff;
        const int rowB = rowA + 16;
        const int dnA  = SCATTER ? sidx[rowA] : 0;
        const int dnB  = SCATTER ? sidx[rowB] : 0;
        #pragma unroll
        for (int n = 0; n < 8; ++n) {
            const int col = n * 16 + r;
            const float vA = cL[n][j] + b2v[n];
            const float vB = cH[n][j] + b2v[n];
            if (NTOUT) {
                __builtin_nontemporal_store(vA, &out[(size_t)rowA * D + col]);
                __builtin_nontemporal_store(vB, &out[(size_t)rowB * D + col]);
            } else {
                out[(size_t)rowA * D + col] = vA;
                out[(size_t)rowB * D + col] = vB;
            }
            if (SCATTER) {
                atomicAdd(&agg[(size_t)dnA * D + col], vA);
                atomicAdd(&agg[(size_t)dnB * D + col], vB);
            }
        }
    }
}

extern "C" void kernel_launch(void* const* d_in, const int* in_sizes, int n_in,
                              void* d_out, int out_size, void* d_ws, size_t ws_size,
                              hipStream_t stream)
{
    const float* nodes0  = (const float*)d_in[0];
    const int*   eidx    = (const int*)d_in[1];
    const float* eattr0  = (const float*)d_in[2];
    const float* msg_w1  = (const float*)d_in[3];
    const float* msg_b1  = (const float*)d_in[4];
    const float* msg_w2  = (const float*)d_in[5];
    const float* msg_b2  = (const float*)d_in[6];
    const float* node_w1 = (const float*)d_in[7];
    const float* node_b1 = (const float*)d_in[8];
    const float* node_w2 = (const float*)d_in[9];
    const float* node_b2 = (const float*)d_in[10];
    const float* fin_w1  = (const float*)d_in[11];
    const float* fin_b1  = (const float*)d_in[12];
    const float* fin_w2  = (const float*)d_in[13];
    const float* fin_b2  = (const float*)d_in[14];

    const int N = in_sizes[0] / D;
    const int E = in_sizes[2] / D;
    const int* src = eidx;          // edge_index[0]
    const int* dst = eidx + E;      // edge_index[1]

    float* nodes_out = (float*)d_out;                      // [N, D] (output 0)
    float* edge_out  = (float*)d_out + (size_t)N * D;      // [E, D] (output 1; layer-1 msg buf)

    // workspace layout: agg [N*D] then fragment-packed weights (~1MB)
    float* agg = (float*)d_ws;
    float* pw  = agg + (size_t)N * D;
    const int KW1M = 3 * D, KW2 = D, KW1N = 2 * D;
    float* msg_w1f[2]  = { pw,                         pw + KW1M * D };
    pw += 2 * KW1M * D;
    float* msg_w2f[2]  = { pw,                         pw + KW2 * D };
    pw += 2 * KW2 * D;
    float* node_w1f[2] = { pw,                         pw + KW1N * D };
    pw += 2 * KW1N * D;
    float* node_w2f[2] = { pw,                         pw + KW2 * D };
    pw += 2 * KW2 * D;
    float* fin_w1f = pw;            pw += KW2 * D;
    float* fin_w2f = pw;

    // pack all weight matrices into fragment order (tiny; runs every call)
    for (int l = 0; l < 2; ++l) {
        pack_wfrag_kernel<<<(KW1M * D + 255) / 256, 256, 0, stream>>>(
            msg_w1 + (size_t)l * KW1M * D, msg_w1f[l], KW1M * D);
        pack_wfrag_kernel<<<(KW2 * D + 255) / 256, 256, 0, stream>>>(
            msg_w2 + (size_t)l * KW2 * D, msg_w2f[l], KW2 * D);
        pack_wfrag_kernel<<<(KW1N * D + 255) / 256, 256, 0, stream>>>(
            node_w1 + (size_t)l * KW1N * D, node_w1f[l], KW1N * D);
        pack_wfrag_kernel<<<(KW2 * D + 255) / 256, 256, 0, stream>>>(
            node_w2 + (size_t)l * KW2 * D, node_w2f[l], KW2 * D);
    }
    pack_wfrag_kernel<<<(KW2 * D + 255) / 256, 256, 0, stream>>>(fin_w1, fin_w1f, KW2 * D);
    pack_wfrag_kernel<<<(KW2 * D + 255) / 256, 256, 0, stream>>>(fin_w2, fin_w2f, KW2 * D);

    const dim3 blk(64);                    // 2 waves, 2 x 32-row tiles per block
    const int egrid = (E + 63) / 64;
    const int ngrid = (N + 63) / 64;

    const float* nodes_cur = nodes0;
    const float* eattr_cur = eattr0;
    for (int l = 0; l < 2; ++l) {
        zero4_kernel<<<512, 256, 0, stream>>>((float4*)agg, (size_t)N * D / 4);

        // edge message MLP: concat(nodes[src], nodes[dst], edge_attr) -> messages
        // (in-place on edge_out at layer 2) + scatter-add into agg[dst].
        mlp2_wmma_kernel<3, true, true, true><<<egrid, blk, 0, stream>>>(
            nodes_cur, src, nodes_cur, dst, eattr_cur, (const int*)nullptr,
            msg_w1f[l], msg_b1 + (size_t)l * D,
            msg_w2f[l], msg_b2 + (size_t)l * D,
            edge_out, dst, agg, E);

        // node update MLP: concat(nodes, agg) -> nodes_out (tables stay RT/L2)
        mlp2_wmma_kernel<2, false, false, false><<<ngrid, blk, 0, stream>>>(
            nodes_cur, (const int*)nullptr, agg, (const int*)nullptr,
            (const float*)nullptr, (const int*)nullptr,
            node_w1f[l], node_b1 + (size_t)l * D,
            node_w2f[l], node_b2 + (size_t)l * D,
            nodes_out, (const int*)nullptr, (float*)nullptr, N);

        nodes_cur = nodes_out;
        eattr_cur = edge_out;
    }

    // final MLP (in-place on nodes_out; last read + final store -> NT both ways)
    mlp2_wmma_kernel<1, false, true, true><<<ngrid, blk, 0, stream>>>(
        nodes_cur, (const int*)nullptr,
        (const float*)nullptr, (const int*)nullptr,
        (const float*)nullptr, (const int*)nullptr,
        fin_w1f, fin_b1, fin_w2f, fin_b2,
        nodes_out, (const int*)nullptr, (float*)nullptr, N);
}